// GraphCapsuleLayer_15066745274552
// MI455X (gfx1250) — compile-verified
//
#include <hip/hip_runtime.h>
#include <hip/hip_bf16.h>
#include <math.h>

typedef float v2f __attribute__((ext_vector_type(2)));
typedef float v8f __attribute__((ext_vector_type(8)));

static constexpr int kNC = 16;   // num capsules
static constexpr int kDC = 16;   // dim capsule
static constexpr int kIC = 32;   // input capsules
static constexpr int kID = 8;    // input dim
static constexpr int kBT = 16;   // batch items per workgroup (WMMA M)
static constexpr float kEPS = 1e-7f;

// LDS layouts (b-index innermost => conflict-free per-lane A reads):
//   lds_u    [kIC*kID][kBT]   u[b,i,j] at (i*8+j)*16 + b
//   lds_coef [kNC][kIC][kBT]  routing coefficient c[b,c,i]
//   lds_brt  [kNC][kIC][kBT]  routing logits b[b,c,i]
//   lds_out  [kBT][kNC][kDC]  s / outputs
//   lds_scr  [8][kBT][kDC]    per-wave u_hat tile scratch

__global__ __launch_bounds__(256, 1)
void capsule_routing_kernel(const float* __restrict__ x,
                            const float* __restrict__ W,
                            float* __restrict__ out) {
    __shared__ float lds_u[kIC * kID * kBT];       //  16 KB
    __shared__ float lds_coef[kNC * kIC * kBT];    //  32 KB
    __shared__ float lds_brt[kNC * kIC * kBT];     //  32 KB
    __shared__ float lds_out[kBT * kNC * kDC];     //  16 KB
    __shared__ float lds_scr[8 * kBT * kDC];       //   8 KB

    const int t     = threadIdx.x;
    const int lane  = t & 31;
    const int wv    = t >> 5;        // wave 0..7; owns capsules 2*wv, 2*wv+1
    const int m     = lane & 15;     // M-row (A) or N-col (B/C) index
    const int khalf = lane >> 4;     // K-pair selector for A/B fragments
    const long bg0  = (long)blockIdx.x * kBT;

    // ---- load u tile (coalesced) + init routing state ----
    {
        const float* xp = x + bg0 * (kIC * kID);
        for (int b = 0; b < kBT; ++b)
            lds_u[t * kBT + b] = xp[b * (kIC * kID) + t];
    }
    for (int k = 0; k < (kNC * kIC * kBT) / 256; ++k) {
        lds_coef[t + 256 * k] = 1.0f / 16.0f;   // softmax(0) over 16 capsules
        lds_brt[t + 256 * k]  = 0.0f;
    }
    __syncthreads();

    for (int iter = 0; iter < 3; ++iter) {
        // ============ S-pass: s[b,c,:] = sum_i coef * (u[b,i,:] @ W[i,:,c]) ============
        // K=256 GEMM per capsule via 64 accumulated v_wmma_f32_16x16x4_f32
        for (int cc = 0; cc < 2; ++cc) {
            const int c = wv * 2 + cc;
            v8f acc = {};
            for (int i = 0; i < kIC; ++i) {
                const float coef = lds_coef[(c * kIC + i) * kBT + m]; // broadcast per half
#pragma unroll
                for (int kb = 0; kb < 2; ++kb) {
                    const int j0 = kb * 4 + khalf * 2;
                    const float a0 = coef * lds_u[(i * kID + j0) * kBT + m];
                    const float a1 = coef * lds_u[(i * kID + j0 + 1) * kBT + m];
                    const float* wp = W + (size_t)(i * kID + j0) * (kNC * kDC) + c * kDC + m;
                    v2f Af = {a0, a1};
                    v2f Bf = {wp[0], wp[kNC * kDC]};
                    acc = __builtin_amdgcn_wmma_f32_16x16x4_f32(
                        false, Af, false, Bf, (short)0, acc, false, false);
                }
            }
#pragma unroll
            for (int r = 0; r < 8; ++r) {
                const int b = r + khalf * 8;
                lds_out[(b * kNC + c) * kDC + m] = acc[r];
            }
        }
        __syncthreads();

        // ============ squash per (b,c) ============
        {
            const int b = t & 15, c = t >> 4;
            float v[kDC];
            float s2 = kEPS;
#pragma unroll
            for (int d = 0; d < kDC; ++d) {
                v[d] = lds_out[(b * kNC + c) * kDC + d];
                s2 += v[d] * v[d];
            }
            const float scale = s2 / ((1.0f + s2) * sqrtf(s2));
#pragma unroll
            for (int d = 0; d < kDC; ++d)
                lds_out[(b * kNC + c) * kDC + d] = v[d] * scale;
        }
        __syncthreads();

        if (iter == 2) break;

        // ============ agreement: b_route[b,c,i] += out[b,c,:]·u_hat[b,c,i,:] ============
        // recompute each (c,i) u_hat tile with WMMA (K=8), dot rows in VALU
        for (int cc = 0; cc < 2; ++cc) {
            const int c = wv * 2 + cc;
            for (int i = 0; i < kIC; ++i) {
                v8f acc = {};
#pragma unroll
                for (int kb = 0; kb < 2; ++kb) {
                    const int j0 = kb * 4 + khalf * 2;
                    const float a0 = lds_u[(i * kID + j0) * kBT + m];
                    const float a1 = lds_u[(i * kID + j0 + 1) * kBT + m];
                    const float* wp = W + (size_t)(i * kID + j0) * (kNC * kDC) + c * kDC + m;
                    v2f Af = {a0, a1};
                    v2f Bf = {wp[0], wp[kNC * kDC]};
                    acc = __builtin_amdgcn_wmma_f32_16x16x4_f32(
                        false, Af, false, Bf, (short)0, acc, false, false);
                }
#pragma unroll
                for (int r = 0; r < 8; ++r) {
                    const int b = r + khalf * 8;
                    lds_scr[(wv * kBT + b) * kDC + m] = acc[r];
                }
                __syncthreads();   // uniform: every wave runs 2*32 tiles
                {
                    const int b = m;   // both lane-halves compute, low half stores
                    float dotv = 0.0f;
#pragma unroll
                    for (int d = 0; d < kDC; ++d)
                        dotv += lds_out[(b * kNC + c) * kDC + d] *
                                lds_scr[(wv * kBT + b) * kDC + d];
                    if (lane < 16)
                        lds_brt[(c * kIC + i) * kBT + b] += dotv;
                }
                __syncthreads();
            }
        }

        // ============ softmax over capsule axis for each (b,i) ============
        for (int rr = 0; rr < 2; ++rr) {
            const int p = t + 256 * rr;   // 512 (b,i) pairs
            const int b = p & 15;
            const int i = p >> 4;
            float vals[kNC];
            float mx = -3.0e38f;
#pragma unroll
            for (int c = 0; c < kNC; ++c) {
                vals[c] = lds_brt[(c * kIC + i) * kBT + b];
                mx = fmaxf(mx, vals[c]);
            }
            float sum = 0.0f;
#pragma unroll
            for (int c = 0; c < kNC; ++c) {
                vals[c] = __expf(vals[c] - mx);
                sum += vals[c];
            }
            const float inv = 1.0f / sum;
#pragma unroll
            for (int c = 0; c < kNC; ++c)
                lds_coef[(c * kIC + i) * kBT + b] = vals[c] * inv;
        }
        __syncthreads();
    }

    // ---- write outputs: out[bg0+b, c, d]; note c*16+d == t => coalesced copy ----
    {
        float* op = out + bg0 * (kNC * kDC);
        for (int b = 0; b < kBT; ++b)
            op[b * (kNC * kDC) + t] = lds_out[b * (kNC * kDC) + t];
    }
}

extern "C" void kernel_launch(void* const* d_in, const int* in_sizes, int n_in,
                              void* d_out, int out_size, void* d_ws, size_t ws_size,
                              hipStream_t stream) {
    (void)n_in; (void)out_size; (void)d_ws; (void)ws_size;
    const float* x = (const float*)d_in[0];   // [64,128,32,8]
    const float* W = (const float*)d_in[1];   // [32,8,256]
    float* out = (float*)d_out;               // [64,128,16,16]
    const int BN = in_sizes[0] / (kIC * kID); // 8192
    dim3 grid(BN / kBT);                      // 512 workgroups
    capsule_routing_kernel<<<grid, 256, 0, stream>>>(x, W, out);
}